// KAN_layer_1108101562900
// MI455X (gfx1250) — compile-verified
//
#include <hip/hip_runtime.h>
#include <math.h>

// KAN layer (OUT_F == IN_F) => elementwise: Y[r,o] = silu(x)*W0[o] + spline5(x)*W1[o] + b[o].
// HBM-bound: 67 MB traffic -> ~2.9us @ 23.3 TB/s. Strategy: double-buffered async
// global->LDS ingest (ASYNCcnt pipeline, NT hints), ds_load_b128 consumption,
// VOPD-friendly minimal VALU per element, NT b128 stores.

typedef __attribute__((ext_vector_type(4))) float v4f;

#define OUTF 1024   // OUT_F == IN_F for this instance
#define RPB  16     // rows per block
#define CH   4      // rows per pipeline stage
#define TPB  256    // 256 threads * 4 cols = 1024 columns

__device__ __forceinline__ float fast_sigmoid(float x) {
    // 1/(1+exp(-x)) via v_exp_f32 + v_rcp_f32 (trans units, off the VALU critical path)
    float e = __builtin_amdgcn_exp2f(x * -1.44269504088896340736f);
    return __builtin_amdgcn_rcpf(1.0f + e);
}

__device__ __forceinline__ void async_copy16(const float* gsrc, unsigned lds_dst) {
    // CDNA5 async global->LDS DMA path, tracked by ASYNCcnt. NT: streamed-once data.
    asm volatile("global_load_async_to_lds_b128 %0, %1, off th:TH_LOAD_NT"
                 :: "v"(lds_dst), "v"(gsrc) : "memory");
}

__global__ __launch_bounds__(TPB)
void kan_async_kernel(const float* __restrict__ X,
                      const float* __restrict__ C,   // (OUTF,5)
                      const float* __restrict__ W,   // (OUTF,2)
                      const float* __restrict__ Bv,  // (OUTF,)
                      float* __restrict__ Y)
{
    __shared__ v4f sh[2][CH][TPB];                   // 32 KB, 2-stage double buffer

    const int tid = threadIdx.x;
    const int col = tid << 2;

    // ---- per-column spline params -> registers (tiny, L2/WGP$-resident) ----
    float base[4][4], slope[4][4], w0[4], w1[4], bb[4];
#pragma unroll
    for (int j = 0; j < 4; ++j) {
        const int o = col + j;
        float cf[5];
#pragma unroll
        for (int g = 0; g < 5; ++g) cf[g] = C[o * 5 + g];
#pragma unroll
        for (int k = 0; k < 4; ++k) { base[j][k] = cf[k]; slope[j][k] = cf[k + 1] - cf[k]; }
        w0[j] = W[o * 2 + 0];
        w1[j] = W[o * 2 + 1];
        bb[j] = Bv[o];
    }

    const size_t row0 = (size_t)blockIdx.x * RPB;
    const float* Xr = X + row0 * OUTF + col;
    float*       Yr = Y + row0 * OUTF + col;

    // ---- prologue: stage 0 in flight ----
#pragma unroll
    for (int u = 0; u < CH; ++u)
        async_copy16(Xr + (size_t)u * OUTF, (unsigned)(uintptr_t)&sh[0][u][tid]);

    for (int rr = 0; rr < RPB; rr += CH) {
        const int s = (rr >> 2) & 1;

        if (rr + CH < RPB) {
            // issue next stage, then release current stage only (async loads
            // complete in order, so ASYNCcnt<=CH means the first CH are done)
#pragma unroll
            for (int u = 0; u < CH; ++u)
                async_copy16(Xr + (size_t)(rr + CH + u) * OUTF,
                             (unsigned)(uintptr_t)&sh[s ^ 1][u][tid]);
            asm volatile("s_wait_asynccnt 0x4" ::: "memory");
        } else {
            asm volatile("s_wait_asynccnt 0x0" ::: "memory");
        }

#pragma unroll
        for (int u = 0; u < CH; ++u) {
            const v4f xv = sh[s][u][tid];            // ds_load_b128
            v4f y;
#pragma unroll
            for (int j = 0; j < 4; ++j) {
                const float x   = xv[j];
                const float sil = x * fast_sigmoid(x);

                const float xc = __builtin_amdgcn_fmed3f(x, -1.0f, 1.0f); // v_med3 clamp
                const float t  = __builtin_fmaf(xc, 2.0f, 2.0f);          // (clip+1)/0.5
                const float kf = fminf(__builtin_floorf(t), 3.0f);
                const float fr = t - kf;

                const bool lt1 = kf < 1.0f, lt2 = kf < 2.0f, lt3 = kf < 3.0f;
                const float b_k = lt2 ? (lt1 ? base[j][0] : base[j][1])
                                      : (lt3 ? base[j][2] : base[j][3]);
                const float s_k = lt2 ? (lt1 ? slope[j][0] : slope[j][1])
                                      : (lt3 ? slope[j][2] : slope[j][3]);
                const float spl = __builtin_fmaf(s_k, fr, b_k);

                y[j] = __builtin_fmaf(sil, w0[j], __builtin_fmaf(spl, w1[j], bb[j]));
            }
            __builtin_nontemporal_store(y, (v4f*)(Yr + (size_t)(rr + u) * OUTF));
        }
    }
}

// Tail kernel for nrows not divisible by RPB (not used for B=8192, kept for generality).
__global__ __launch_bounds__(TPB)
void kan_tail_kernel(const float* __restrict__ X, const float* __restrict__ C,
                     const float* __restrict__ W, const float* __restrict__ Bv,
                     float* __restrict__ Y, int row_start, int nrows)
{
    const int qpr  = OUTF / 4;
    const int idx  = blockIdx.x * blockDim.x + threadIdx.x;
    const int rem  = (nrows - row_start) * qpr;
    if (idx >= rem) return;
    const int r   = row_start + idx / qpr;
    const int col = (idx % qpr) * 4;

    v4f xv = *(const v4f*)(X + (size_t)r * OUTF + col);
    v4f y;
#pragma unroll
    for (int j = 0; j < 4; ++j) {
        const int o = col + j;
        float cf[5];
#pragma unroll
        for (int g = 0; g < 5; ++g) cf[g] = C[o * 5 + g];
        const float x   = xv[j];
        const float sil = x * fast_sigmoid(x);
        const float xc  = __builtin_amdgcn_fmed3f(x, -1.0f, 1.0f);
        const float t   = __builtin_fmaf(xc, 2.0f, 2.0f);
        const float kf  = fminf(__builtin_floorf(t), 3.0f);
        const float fr  = t - kf;
        const bool lt1 = kf < 1.0f, lt2 = kf < 2.0f, lt3 = kf < 3.0f;
        const float c0 = lt2 ? (lt1 ? cf[0] : cf[1]) : (lt3 ? cf[2] : cf[3]);
        const float c1 = lt2 ? (lt1 ? cf[1] : cf[2]) : (lt3 ? cf[3] : cf[4]);
        const float spl = __builtin_fmaf(c1 - c0, fr, c0);
        y[j] = __builtin_fmaf(sil, W[o * 2 + 0], __builtin_fmaf(spl, W[o * 2 + 1], Bv[o]));
    }
    __builtin_nontemporal_store(y, (v4f*)(Y + (size_t)r * OUTF + col));
}

extern "C" void kernel_launch(void* const* d_in, const int* in_sizes, int n_in,
                              void* d_out, int out_size, void* d_ws, size_t ws_size,
                              hipStream_t stream) {
    const float* X  = (const float*)d_in[0];   // (B, 1024)
    const float* C  = (const float*)d_in[1];   // (1024, 5)
    const float* W  = (const float*)d_in[2];   // (1024, 2)
    const float* b  = (const float*)d_in[3];   // (1024,)
    float* Y = (float*)d_out;

    const int nrows = in_sizes[0] / OUTF;      // B = 8192
    const int full  = nrows / RPB;             // 512 full tiles
    if (full > 0)
        kan_async_kernel<<<full, TPB, 0, stream>>>(X, C, W, b, Y);

    const int row_start = full * RPB;
    const int rem_rows  = nrows - row_start;   // 0 for B=8192
    if (rem_rows > 0) {
        const int work = rem_rows * (OUTF / 4);
        kan_tail_kernel<<<(work + TPB - 1) / TPB, TPB, 0, stream>>>(X, C, W, b, Y,
                                                                    row_start, nrows);
    }
}